// VSSBlock_84215718740565
// MI455X (gfx1250) — compile-verified
//
#include <hip/hip_runtime.h>

// Problem constants (fixed by the reference)
#define Bq   2
#define Hq   128
#define Wq   128
#define Cq   96
#define Lq   (Hq*Wq)        // 16384
#define BLq  (Bq*Lq)        // 32768
#define DIq  144
#define Rq   6
#define Nq   16
#define XPq  (Rq + 2*Nq)    // 38 valid cols of x_proj output
#define XPAD 64             // padded stride (4 full WMMA tiles)

typedef float v2f __attribute__((ext_vector_type(2)));
typedef float v8f __attribute__((ext_vector_type(8)));

__device__ __forceinline__ float sigmoidf_(float x) { return 1.f / (1.f + __expf(-x)); }

// ---------------------------------------------------------------------------
// LayerNorm over last dim (C=96): one wave32 per row, 3 columns per lane.
// ---------------------------------------------------------------------------
__global__ void k_layernorm(const float* __restrict__ x, const float* __restrict__ g,
                            const float* __restrict__ bt, float* __restrict__ y, float eps)
{
    int lane = threadIdx.x & 31;
    int wv   = threadIdx.x >> 5;
    int row  = blockIdx.x * 4 + wv;
    const float* xr = x + (size_t)row * Cq;
    float v0 = xr[lane], v1 = xr[lane + 32], v2 = xr[lane + 64];
    float s = v0 + v1 + v2;
#pragma unroll
    for (int m = 16; m >= 1; m >>= 1) s += __shfl_xor(s, m, 32);
    float mu = s * (1.f / Cq);
    float d0 = v0 - mu, d1 = v1 - mu, d2 = v2 - mu;
    float q = d0 * d0 + d1 * d1 + d2 * d2;
#pragma unroll
    for (int m = 16; m >= 1; m >>= 1) q += __shfl_xor(q, m, 32);
    float r = rsqrtf(q * (1.f / Cq) + eps);
    float* yr = y + (size_t)row * Cq;
    yr[lane]      = d0 * r * g[lane]      + bt[lane];
    yr[lane + 32] = d1 * r * g[lane + 32] + bt[lane + 32];
    yr[lane + 64] = d2 * r * g[lane + 64] + bt[lane + 64];
}

// ---------------------------------------------------------------------------
// Skip-term init: out = 4 * skip_scale[c] * xn  (sum of 4 un-permuted skips)
// ---------------------------------------------------------------------------
__global__ void k_init_out(const float* __restrict__ xn, const float* __restrict__ ss,
                           float* __restrict__ out)
{
    int idx = blockIdx.x * 256 + threadIdx.x;      // BL*C
    int c = idx % Cq;
    out[idx] = 4.f * ss[c] * xn[idx];
}

// ---------------------------------------------------------------------------
// Zero-pad x_proj_w (38 x 144) into (64 x 144) so all GEMMs are maskless.
// ---------------------------------------------------------------------------
__global__ void k_padw(const float* __restrict__ src, float* __restrict__ dst)
{
    int i = blockIdx.x * 256 + threadIdx.x;        // XPAD*DI = 9216
    int r = i / DIq;
    dst[i] = (r < XPq) ? src[i] : 0.f;
}

// ---------------------------------------------------------------------------
// f32 WMMA GEMM, two 16-wide N-tiles per wave (A fragment reused across both),
// software-pipelined one k-step deep so fragment loads for step k+1 are in
// flight while the two WMMAs of step k execute (distinct regs -> no re-wait).
//   D[m, n] = sum_k A[row(m), k] * Wt[n, k]
//  - optional gather of A rows (scan permutation) and scatter-add of D rows
//    (inverse permutation folded in as out[b, scan[r]] += yk[b, r]).
//  - Ncols must be a multiple of 32; partial-N handled by zero-padded weights.
// f32 WMMA fragment layout (ISA 7.12.2): A/B are 2 VGPRs, half-wave 0 holds
// K={k0,k0+1}, half-wave 1 holds K={k0+2,k0+3}; C/D 8 VGPRs, M = v + 8*half.
// ---------------------------------------------------------------------------
template <int K>
__global__ void k_gemm_wmma(const float* __restrict__ A, int lda,
                            const float* __restrict__ Wt, int ldw,
                            float* __restrict__ D, int ldd,
                            int Ncols,
                            const int* __restrict__ gatherIdx,
                            const int* __restrict__ scatterIdx,
                            int accumulate)
{
    const int lane = threadIdx.x & 31;
    const int wv   = threadIdx.x >> 5;
    const int m0   = blockIdx.x * 16;
    const int n0   = (blockIdx.y * 2 + wv) * 32;   // 2 tiles (32 cols) per wave
    if (n0 >= Ncols) return;                       // uniform per wave

    int arow = m0 + (lane & 15);
    int grow = arow;
    if (gatherIdx) grow = (arow & ~(Lq - 1)) + gatherIdx[arow & (Lq - 1)];
    const float* ap  = A + (size_t)grow * lda;

    const float* wp0 = Wt + (size_t)(n0 + (lane & 15)) * ldw;
    const float* wp1 = wp0 + (size_t)16 * ldw;

    const int koff = (lane >> 4) << 1;             // 0 for lanes 0-15, 2 for 16-31
    v8f acc0 = {}, acc1 = {};

    // prologue
    v2f av, b0, b1;
    av.x = ap[koff];  av.y = ap[koff + 1];
    b0.x = wp0[koff]; b0.y = wp0[koff + 1];
    b1.x = wp1[koff]; b1.y = wp1[koff + 1];

#pragma unroll
    for (int k0 = 0; k0 < K - 4; k0 += 4) {
        v2f avn, b0n, b1n;                         // next-step fragments
        avn.x = ap[k0 + 4 + koff];  avn.y = ap[k0 + 4 + koff + 1];
        b0n.x = wp0[k0 + 4 + koff]; b0n.y = wp0[k0 + 4 + koff + 1];
        b1n.x = wp1[k0 + 4 + koff]; b1n.y = wp1[k0 + 4 + koff + 1];
        acc0 = __builtin_amdgcn_wmma_f32_16x16x4_f32(false, av, false, b0,
                                                     (short)0, acc0, false, false);
        acc1 = __builtin_amdgcn_wmma_f32_16x16x4_f32(false, av, false, b1,
                                                     (short)0, acc1, false, false);
        av = avn; b0 = b0n; b1 = b1n;
    }
    acc0 = __builtin_amdgcn_wmma_f32_16x16x4_f32(false, av, false, b0,
                                                 (short)0, acc0, false, false);
    acc1 = __builtin_amdgcn_wmma_f32_16x16x4_f32(false, av, false, b1,
                                                 (short)0, acc1, false, false);

    int rbase = m0 + ((lane >> 4) << 3);
    int c0 = n0 + (lane & 15);
#pragma unroll
    for (int v = 0; v < 8; ++v) {
        int row  = rbase + v;
        int drow = row;
        if (scatterIdx) drow = (row & ~(Lq - 1)) + scatterIdx[row & (Lq - 1)];
        float* dp = D + (size_t)drow * ldd;
        if (accumulate) { dp[c0] += acc0[v]; dp[c0 + 16] += acc1[v]; }
        else            { dp[c0]  = acc0[v]; dp[c0 + 16]  = acc1[v]; }
    }
}

// ---------------------------------------------------------------------------
// Causal depthwise conv(k=3) along the scanned sequence + SiLU.
// ---------------------------------------------------------------------------
__global__ void k_conv1d_silu(const float* __restrict__ xz, const float* __restrict__ cw,
                              const float* __restrict__ cb, float* __restrict__ xc)
{
    int idx = blockIdx.x * 256 + threadIdx.x;      // BL*DI
    int d = idx % DIq;
    int m = idx / DIq;
    int r = m & (Lq - 1);
    int base = m - r;                              // b*L
    float acc = cb[d];
#pragma unroll
    for (int t = 0; t < 3; ++t) {
        int rr = r - 2 + t;
        if (rr >= 0) acc += xz[(size_t)(base + rr) * (2 * DIq) + d] * cw[d * 3 + t];
    }
    xc[(size_t)m * DIq + d] = acc * sigmoidf_(acc);
}

// ---------------------------------------------------------------------------
// dt = softplus(xdbl[:, :6] @ dt_w^T + dt_b)   (xdbl has padded stride 64)
// ---------------------------------------------------------------------------
__global__ void k_dtproj(const float* __restrict__ xdbl, const float* __restrict__ dtw,
                         const float* __restrict__ dtb, float* __restrict__ dt)
{
    int idx = blockIdx.x * 256 + threadIdx.x;      // BL*DI
    int d = idx % DIq;
    int m = idx / DIq;
    const float* xr = xdbl + (size_t)m * XPAD;
    float a = dtb[d];
#pragma unroll
    for (int j = 0; j < Rq; ++j) a += xr[j] * dtw[d * Rq + j];
    dt[(size_t)m * DIq + d] = (a > 20.f) ? a : log1pf(__expf(a));
}

// ---------------------------------------------------------------------------
// Selective scan: lanes = 16 SSM states, 2 channels per wave (wave32).
// Critical path per step is exp-term (independent of h) + 2 FMAs on h.
// Speculative prefetch 8 rows ahead keeps the streamed operands in flight
// while the recurrence chains (global_prefetch_b8; OOB prefetch is dropped).
// ---------------------------------------------------------------------------
__global__ void k_scan(const float* __restrict__ dt, const float* __restrict__ xc,
                       const float* __restrict__ xdbl, const float* __restrict__ A_log,
                       float* __restrict__ ysum)
{
    int lane = threadIdx.x & 31;
    int wv   = threadIdx.x >> 5;                   // 0..7
    int half = lane >> 4;
    int n    = lane & 15;
    int b    = blockIdx.x / (DIq / 16);            // grid.x = B * 9
    int dgrp = blockIdx.x % (DIq / 16);
    int d    = dgrp * 16 + wv * 2 + half;

    float a = -expf(A_log[d * Nq + n]);
    float h = 0.f;
    size_t rowBase = (size_t)b * Lq;
    for (int r = 0; r < Lq; ++r) {
        size_t m = rowBase + r;
        __builtin_prefetch(dt   + (m + 8) * DIq  + d, 0, 1);
        __builtin_prefetch(xc   + (m + 8) * DIq  + d, 0, 1);
        __builtin_prefetch(xdbl + (m + 8) * XPAD + Rq + n, 0, 1);
        float dtv = dt[m * DIq + d];               // broadcast within half-wave
        float xv  = xc[m * DIq + d];
        float Bv  = xdbl[m * XPAD + Rq + n];
        float Cv  = xdbl[m * XPAD + Rq + Nq + n];
        h = __expf(dtv * a) * h + (dtv * Bv) * xv;
        float p = h * Cv;
        p += __shfl_xor(p, 1, 32);
        p += __shfl_xor(p, 2, 32);
        p += __shfl_xor(p, 4, 32);
        p += __shfl_xor(p, 8, 32);                 // reduce over 16 states
        if (n == 0) ysum[m * DIq + d] = p;
    }
}

// ---------------------------------------------------------------------------
// g = (ys + xc*Dp) * silu(z)
// ---------------------------------------------------------------------------
__global__ void k_gate(const float* __restrict__ ys, const float* __restrict__ xc,
                       const float* __restrict__ xz, const float* __restrict__ Dp,
                       float* __restrict__ g)
{
    int idx = blockIdx.x * 256 + threadIdx.x;      // BL*DI  (== m*DI + d)
    int d = idx % DIq;
    int m = idx / DIq;
    float z = xz[(size_t)m * (2 * DIq) + DIq + d];
    float y = ys[idx] + xc[idx] * Dp[d];
    g[idx] = y * (z * sigmoidf_(z));
}

// ---------------------------------------------------------------------------
// 3x3 conv 96->32 (SAME) + exact GELU.  Pixel address broadcast across lanes,
// weights coalesced across co.
// ---------------------------------------------------------------------------
__global__ void k_conv1_gelu(const float* __restrict__ x, const float* __restrict__ w,
                             const float* __restrict__ bias, float* __restrict__ u)
{
    int idx = blockIdx.x * 256 + threadIdx.x;      // B*H*W*32
    int co = idx & 31;
    int wq = (idx >> 5) & (Wq - 1);
    int hq = (idx >> 12) & (Hq - 1);
    int b  = idx >> 19;
    float acc = bias[co];
    for (int kh = 0; kh < 3; ++kh) {
        int hh = hq + kh - 1; if (hh < 0 || hh >= Hq) continue;
        for (int kw = 0; kw < 3; ++kw) {
            int ww = wq + kw - 1; if (ww < 0 || ww >= Wq) continue;
            const float* px = x + (size_t)(((b * Hq + hh) * Wq) + ww) * Cq;
            const float* pw = w + (size_t)((kh * 3 + kw) * Cq) * 32 + co;
            for (int ci = 0; ci < Cq; ++ci) acc += px[ci] * pw[ci * 32];
        }
    }
    u[idx] = 0.5f * acc * (1.f + erff(acc * 0.70710678118f));
}

// ---------------------------------------------------------------------------
// 3x3 conv 32->96 (SAME) + bias -> t
// ---------------------------------------------------------------------------
__global__ void k_conv2(const float* __restrict__ u, const float* __restrict__ w,
                        const float* __restrict__ bias, float* __restrict__ t)
{
    int idx = blockIdx.x * 256 + threadIdx.x;      // B*H*W*96
    int c = idx % Cq;
    int p = idx / Cq;
    int wq = p & (Wq - 1);
    int hq = (p >> 7) & (Hq - 1);
    int b  = p >> 14;
    float acc = bias[c];
    for (int kh = 0; kh < 3; ++kh) {
        int hh = hq + kh - 1; if (hh < 0 || hh >= Hq) continue;
        for (int kw = 0; kw < 3; ++kw) {
            int ww = wq + kw - 1; if (ww < 0 || ww >= Wq) continue;
            const float* px = u + (size_t)(((b * Hq + hh) * Wq) + ww) * 32;
            const float* pw = w + (size_t)((kh * 3 + kw) * 32) * Cq + c;
#pragma unroll
            for (int ci = 0; ci < 32; ++ci) acc += px[ci] * pw[ci * Cq];
        }
    }
    t[idx] = acc;
}

// ---------------------------------------------------------------------------
// Global mean pool over H*W per (b, c)
// ---------------------------------------------------------------------------
__global__ void k_pool(const float* __restrict__ t, float* __restrict__ pooled)
{
    __shared__ float sm[256];
    int c = blockIdx.x % Cq;
    int b = blockIdx.x / Cq;
    float s = 0.f;
    for (int i = threadIdx.x; i < Lq; i += 256)
        s += t[(size_t)(b * Lq + i) * Cq + c];
    sm[threadIdx.x] = s;
    __syncthreads();
    for (int st = 128; st > 0; st >>= 1) {
        if (threadIdx.x < st) sm[threadIdx.x] += sm[threadIdx.x + st];
        __syncthreads();
    }
    if (threadIdx.x == 0) pooled[blockIdx.x] = sm[0] * (1.f / Lq);
}

// ---------------------------------------------------------------------------
// Channel attention: 1x1 96->3 relu -> 1x1 3->96 sigmoid
// ---------------------------------------------------------------------------
__global__ void k_attn(const float* __restrict__ pooled,
                       const float* __restrict__ w1, const float* __restrict__ b1,
                       const float* __restrict__ w2, const float* __restrict__ b2,
                       float* __restrict__ amap)
{
    __shared__ float mid[3];
    int b = blockIdx.x;
    int c = threadIdx.x;                           // 0..95
    if (c < 3) {
        float a = b1[c];
        for (int ci = 0; ci < Cq; ++ci) a += pooled[b * Cq + ci] * w1[ci * 3 + c];
        mid[c] = fmaxf(a, 0.f);
    }
    __syncthreads();
    float a = b2[c];
#pragma unroll
    for (int j = 0; j < 3; ++j) a += mid[j] * w2[j * Cq + c];
    amap[b * Cq + c] = sigmoidf_(a);
}

// ---------------------------------------------------------------------------
// Final combine: out = x*skip_scale2 + t*a
// ---------------------------------------------------------------------------
__global__ void k_final(const float* __restrict__ x, const float* __restrict__ t,
                        const float* __restrict__ ss2, const float* __restrict__ amap,
                        float* __restrict__ out)
{
    int idx = blockIdx.x * 256 + threadIdx.x;      // BL*C
    int c = idx % Cq;
    int b = idx / (Lq * Cq);
    out[idx] = x[idx] * ss2[c] + t[idx] * amap[b * Cq + c];
}

// ===========================================================================
extern "C" void kernel_launch(void* const* d_in, const int* in_sizes, int n_in,
                              void* d_out, int out_size, void* d_ws, size_t ws_size,
                              hipStream_t stream)
{
    (void)in_sizes; (void)n_in; (void)out_size; (void)ws_size;

    const float* input   = (const float*)d_in[0];
    const int*   scanIds = (const int*)  d_in[1];   // (4, L)
    /* d_in[2] inv_ids: not needed (scatter uses scan_ids identity) */
    const float* ln1_g   = (const float*)d_in[3];
    const float* ln1_b   = (const float*)d_in[4];
    const float* inp_w   = (const float*)d_in[5];   // (288, 96)
    const float* conv_w  = (const float*)d_in[6];   // (144, 3)
    const float* conv_b  = (const float*)d_in[7];
    const float* xp_w    = (const float*)d_in[8];   // (38, 144)
    const float* dt_w    = (const float*)d_in[9];   // (144, 6)
    const float* dt_b    = (const float*)d_in[10];
    const float* A_log   = (const float*)d_in[11];  // (144, 16)
    const float* Dp      = (const float*)d_in[12];
    const float* outp_w  = (const float*)d_in[13];  // (96, 144)
    const float* ss      = (const float*)d_in[14];
    const float* ln2_g   = (const float*)d_in[15];
    const float* ln2_b   = (const float*)d_in[16];
    const float* cab_w1  = (const float*)d_in[17];  // (3,3,96,32)
    const float* cab_b1  = (const float*)d_in[18];
    const float* cab_w2  = (const float*)d_in[19];  // (3,3,32,96)
    const float* cab_b2  = (const float*)d_in[20];
    const float* ca_w1   = (const float*)d_in[21];  // (1,1,96,3)
    const float* ca_b1   = (const float*)d_in[22];
    const float* ca_w2   = (const float*)d_in[23];  // (1,1,3,96)
    const float* ca_b2   = (const float*)d_in[24];
    const float* ss2     = (const float*)d_in[25];
    float* dout = (float*)d_out;

    // workspace carve-up (floats), ~170 MB peak; buffers reused across the 4 dirs
    float* W = (float*)d_ws;
    size_t off = 0;
    auto alloc = [&](size_t n) { float* p = W + off; off += (n + 63) & ~(size_t)63; return p; };
    float* xn   = alloc((size_t)BLq * Cq);
    float* xz   = alloc((size_t)BLq * 2 * DIq);
    float* xc   = alloc((size_t)BLq * DIq);
    float* xdbl = alloc((size_t)BLq * XPAD);
    float* dtb  = alloc((size_t)BLq * DIq);
    float* ys   = alloc((size_t)BLq * DIq);
    float* gb   = alloc((size_t)BLq * DIq);
    float* xout = alloc((size_t)BLq * Cq);
    float* xln  = alloc((size_t)BLq * Cq);
    float* ub   = alloc((size_t)BLq * 32);
    float* tb   = alloc((size_t)BLq * Cq);
    float* pool = alloc(Bq * Cq);
    float* amap = alloc(Bq * Cq);
    float* wpad = alloc((size_t)XPAD * DIq);

    // 0) zero-pad x_proj weights (once) so all GEMMs are maskless
    k_padw<<<(XPAD * DIq) / 256, 256, 0, stream>>>(xp_w, wpad);
    // 1) LN1
    k_layernorm<<<BLq / 4, 128, 0, stream>>>(input, ln1_g, ln1_b, xn, 1e-6f);
    // 2) out = 4 * skip_scale * xn   (analytic sum of the 4 gathered skips)
    k_init_out<<<(BLq * Cq) / 256, 256, 0, stream>>>(xn, ss, xout);

    // 3) four directional Mamba branches (sequential, buffers reused)
    for (int k = 0; k < 4; ++k) {
        const int* perm = scanIds + (size_t)k * Lq;
        // in_proj: (BL,96) x (96,288), gather rows through scan permutation
        k_gemm_wmma<Cq><<<dim3(BLq / 16, 5), 64, 0, stream>>>(
            xn, Cq, inp_w, Cq, xz, 2 * DIq, 2 * DIq, perm, nullptr, 0);
        // depthwise causal conv3 + SiLU
        k_conv1d_silu<<<(BLq * DIq) / 256, 256, 0, stream>>>(xz, conv_w, conv_b, xc);
        // x_proj: (BL,144) x (144,64-padded)
        k_gemm_wmma<DIq><<<dim3(BLq / 16, 1), 64, 0, stream>>>(
            xc, DIq, wpad, DIq, xdbl, XPAD, XPAD, nullptr, nullptr, 0);
        // dt projection + softplus
        k_dtproj<<<(BLq * DIq) / 256, 256, 0, stream>>>(xdbl, dt_w, dt_b, dtb);
        // selective scan
        k_scan<<<Bq * (DIq / 16), 256, 0, stream>>>(dtb, xc, xdbl, A_log, ys);
        // gate: (ys + xc*Dp) * silu(z)
        k_gate<<<(BLq * DIq) / 256, 256, 0, stream>>>(ys, xc, xz, Dp, gb);
        // out_proj: (BL,144) x (144,96), scatter-add out[b, scan[r]] += row r
        k_gemm_wmma<DIq><<<dim3(BLq / 16, 2), 64, 0, stream>>>(
            gb, DIq, outp_w, DIq, xout, Cq, Cq, nullptr, perm, 1);
    }

    // 4) spatial branch
    k_layernorm<<<BLq / 4, 128, 0, stream>>>(xout, ln2_g, ln2_b, xln, 1e-5f);
    k_conv1_gelu<<<(BLq * 32) / 256, 256, 0, stream>>>(xln, cab_w1, cab_b1, ub);
    k_conv2<<<(BLq * Cq) / 256, 256, 0, stream>>>(ub, cab_w2, cab_b2, tb);
    k_pool<<<Bq * Cq, 256, 0, stream>>>(tb, pool);
    k_attn<<<Bq, Cq, 0, stream>>>(pool, ca_w1, ca_b1, ca_w2, ca_b2, amap);
    k_final<<<(BLq * Cq) / 256, 256, 0, stream>>>(xout, tb, ss2, amap, dout);
}